// Quantize_separate_22892175687682
// MI455X (gfx1250) — compile-verified
//
#include <hip/hip_runtime.h>
#include <hip/hip_bf16.h>

// Problem constants (from reference)
#define GROUPS   8
#define DIMZ     512
#define DG       64        // DIM / GROUPS
#define KCODES   512       // N_EMBED
#define TTOT     32768     // B*N = 8*4096
#define ROWS     (TTOT * GROUPS)          // 262144 rows of z_q / z_e
#define ZQ_ELEMS (ROWS * DG)              // 16777216
#define LDS_STRIDE 68                     // padded row stride (dwords) -> conflict-free b64 reads
#define LDS_BYTES  (KCODES * LDS_STRIDE * 4 + KCODES * 4)   // E tile + e_sq = 141312 B
// d_out layout: [0, ZQ_ELEMS) = z_q, [ZQ_ELEMS] = diff, [ZQ_ELEMS+1, +ROWS) = indices (as float)

typedef float v2f __attribute__((ext_vector_type(2)));
typedef float v8f __attribute__((ext_vector_type(8)));

// ---------------------------------------------------------------------------
// Kernel 0: e_sq[g*K + k] = sum_d E[g,k,d]^2   (4096 rows x 64)
// ---------------------------------------------------------------------------
__global__ __launch_bounds__(256) void esq_kernel(const float* __restrict__ ew,
                                                  float* __restrict__ esq) {
    int k = blockIdx.x * 256 + threadIdx.x;      // 0..4095
    const float* row = ew + (size_t)k * DG;
    float s = 0.0f;
#pragma unroll
    for (int d = 0; d < DG; ++d) s = __builtin_fmaf(row[d], row[d], s);
    esq[k] = s;
}

// ---------------------------------------------------------------------------
// Kernel 1: per-(token-tile, group) argmin over K=512 codes using f32 WMMA.
// Workgroup (8 waves) stages the group codebook E_g (512x64 f32, stride-68
// padded -> 32 distinct bank pairs per B fetch) + e_sq into LDS once. Each
// wave computes 16 token rows: A (16x64 of X) in regs; k-tile loop is
// double-buffered (bA/bB) so next tile's ds_loads overlap the current
// 16-WMMA chain; accumulator split into two 8-deep chains for ILP.
// ---------------------------------------------------------------------------
__global__ __launch_bounds__(256) void argmin_kernel(const float* __restrict__ z,
                                                     const float* __restrict__ ew,
                                                     const float* __restrict__ esq,
                                                     float* __restrict__ ind_f) {
    extern __shared__ float smem[];                  // [KCODES*LDS_STRIDE] E, then [KCODES] e_sq
    float* smem_esq = smem + KCODES * LDS_STRIDE;

    const int lane = threadIdx.x & 31;
    const int wave = threadIdx.x >> 5;
    const int g    = blockIdx.y;
    const int t0   = (blockIdx.x * 8 + wave) * 16;   // 16-row token tile
    const int half = lane >> 4;                      // 0 or 1
    const int l16  = lane & 15;

    const float* eg   = ew  + (size_t)g * KCODES * DG;
    const float* esqg = esq + g * KCODES;

    // --- Stage E_g into LDS (coalesced float4 copy; one pass per workgroup) ---
#pragma unroll
    for (int it = 0; it < 32; ++it) {
        int idx4 = it * 256 + threadIdx.x;           // 0..8191
        int row  = idx4 >> 4;
        int col4 = idx4 & 15;
        *(float4*)(smem + row * LDS_STRIDE + col4 * 4) =
            *(const float4*)(eg + (size_t)row * DG + col4 * 4);
    }
    // --- Stage e_sq (512 floats) ---
    smem_esq[threadIdx.x]       = esqg[threadIdx.x];
    smem_esq[threadIdx.x + 256] = esqg[threadIdx.x + 256];

    // --- A tile into registers while staging settles ---
    // A-matrix 16x4 f32 layout: lanes 0-15 -> K={0,1} in comps, lanes 16-31 -> K={2,3}
    const float* xrow = z + (size_t)(t0 + l16) * DIMZ + g * DG + 2 * half;
    v2f a[16];
#pragma unroll
    for (int j = 0; j < 16; ++j)
        a[j] = *(const v2f*)(xrow + 4 * j);

    float best[8];
    int   bk[8];
#pragma unroll
    for (int i = 0; i < 8; ++i) { best[i] = 3.4e38f; bk[i] = 0; }

    __syncthreads();

    // B-matrix 4x16 f32 layout: N = lane&15, K = 2*half + comp.
#define LOAD_TILE(BUF, KT) do {                                               \
        const float* brow_ = smem + ((KT) * 16 + l16) * LDS_STRIDE + 2 * half;\
        _Pragma("unroll")                                                     \
        for (int j = 0; j < 16; ++j) BUF[j] = *(const v2f*)(brow_ + 4 * j);   \
    } while (0)

    // C/D layout: VGPR i, lanes 0-15 -> M=i; lanes 16-31 -> M=8+i; N = lane&15.
#define COMPUTE_TILE(BUF, KT) do {                                            \
        v8f c0 = {};                                                          \
        v8f c1 = {};                                                          \
        _Pragma("unroll")                                                     \
        for (int j = 0; j < 8; ++j) {                                         \
            c0 = __builtin_amdgcn_wmma_f32_16x16x4_f32(                       \
                     false, a[j],     false, BUF[j],     (short)0, c0, false, false); \
            c1 = __builtin_amdgcn_wmma_f32_16x16x4_f32(                       \
                     false, a[j + 8], false, BUF[j + 8], (short)0, c1, false, false); \
        }                                                                     \
        const int   k_  = (KT) * 16 + l16;                                    \
        const float es_ = smem_esq[k_];                                       \
        _Pragma("unroll")                                                     \
        for (int i = 0; i < 8; ++i) {                                         \
            float s_ = __builtin_fmaf(-2.0f, c0[i] + c1[i], es_);             \
            if (s_ < best[i]) { best[i] = s_; bk[i] = k_; }                   \
        }                                                                     \
    } while (0)

    v2f bA[16], bB[16];
    LOAD_TILE(bA, 0);
#pragma unroll 1
    for (int kt = 0; kt < 32; kt += 2) {
        LOAD_TILE(bB, kt + 1);           // overlaps bA's WMMA chain
        COMPUTE_TILE(bA, kt);
        if (kt + 2 < 32) LOAD_TILE(bA, kt + 2);  // overlaps bB's WMMA chain
        COMPUTE_TILE(bB, kt + 1);
    }
#undef LOAD_TILE
#undef COMPUTE_TILE

    // Min-reduce each row across its 16 lanes (xor offsets < 16 stay in the half).
#pragma unroll
    for (int off = 1; off < 16; off <<= 1) {
#pragma unroll
        for (int i = 0; i < 8; ++i) {
            float ob = __shfl_xor(best[i], off, 32);
            int   ok = __shfl_xor(bk[i],   off, 32);
            if (ob < best[i] || (ob == best[i] && ok < bk[i])) {
                best[i] = ob; bk[i] = ok;
            }
        }
    }
    if (l16 == 0) {
#pragma unroll
        for (int i = 0; i < 8; ++i) {
            int t = t0 + i + 8 * half;
            ind_f[(size_t)g * TTOT + t] = (float)bk[i];  // r = g*T + t (ind_cat order)
        }
    }
}

// ---------------------------------------------------------------------------
// Kernel 2: gather z_q rows, write output, deterministic per-block sq-diff sums.
// row r: z_q[r] = embed[idx[r]] (chunk-0 gather, faithful); z_e[r] = z + 64*r.
// ---------------------------------------------------------------------------
__global__ __launch_bounds__(256) void gather_kernel(const float* __restrict__ z,
                                                     const float* __restrict__ ew,
                                                     const float* __restrict__ ind_f,
                                                     float* __restrict__ zq_out,
                                                     float* __restrict__ block_sums) {
    __shared__ float sdata[8];
    const int r   = blockIdx.x * 256 + threadIdx.x;   // 0..ROWS-1
    const int idx = (int)ind_f[r];
    const float4* q = (const float4*)(ew + (size_t)idx * DG);
    const float4* e = (const float4*)(z  + (size_t)r   * DG);
    float4*       o = (float4*)(zq_out + (size_t)r * DG);
    float acc = 0.0f;
#pragma unroll
    for (int j = 0; j < 16; ++j) {
        float4 qv = q[j];
        float4 ev = e[j];
        o[j] = qv;
        float dx = qv.x - ev.x, dy = qv.y - ev.y;
        float dz = qv.z - ev.z, dw = qv.w - ev.w;
        acc += dx * dx + dy * dy + dz * dz + dw * dw;
    }
    // fixed-order wave reduce
#pragma unroll
    for (int off = 16; off > 0; off >>= 1) acc += __shfl_down(acc, off, 32);
    if ((threadIdx.x & 31) == 0) sdata[threadIdx.x >> 5] = acc;
    __syncthreads();
    if (threadIdx.x == 0) {
        float s = 0.0f;
#pragma unroll
        for (int w = 0; w < 8; ++w) s += sdata[w];
        block_sums[blockIdx.x] = s;
    }
}

// ---------------------------------------------------------------------------
// Kernel 3: deterministic reduce of 1024 block sums; diff = 12.5 * mean.
// ---------------------------------------------------------------------------
__global__ __launch_bounds__(256) void finalize_kernel(const float* __restrict__ block_sums,
                                                       float* __restrict__ diff_out) {
    __shared__ float sdata[256];
    float s = 0.0f;
    for (int i = threadIdx.x; i < 1024; i += 256) s += block_sums[i];
    sdata[threadIdx.x] = s;
    __syncthreads();
    for (int stride = 128; stride > 0; stride >>= 1) {
        if (threadIdx.x < stride) sdata[threadIdx.x] += sdata[threadIdx.x + stride];
        __syncthreads();
    }
    if (threadIdx.x == 0)
        diff_out[0] = sdata[0] * (12.5f / (float)ZQ_ELEMS);
}

// ---------------------------------------------------------------------------
extern "C" void kernel_launch(void* const* d_in, const int* in_sizes, int n_in,
                              void* d_out, int out_size, void* d_ws, size_t ws_size,
                              hipStream_t stream) {
    const float* z  = (const float*)d_in[0];          // [8,4096,512]
    const float* ew = (const float*)d_in[1];          // [4096,64]
    float* out = (float*)d_out;

    float* zq     = out;                               // [ROWS*64]
    float* diff   = out + ZQ_ELEMS;                    // [1]
    float* ind_f  = out + ZQ_ELEMS + 1;                // [ROWS] indices as float

    float* esq        = (float*)d_ws;                  // 4096 floats
    float* block_sums = esq + GROUPS * KCODES;         // 1024 floats

    // 0) codebook squared norms
    esq_kernel<<<dim3(GROUPS * KCODES / 256), dim3(256), 0, stream>>>(ew, esq);

    // 1) WMMA distance GEMM + argmin -> indices (group-major order)
    argmin_kernel<<<dim3(TTOT / 128, GROUPS), dim3(256), LDS_BYTES, stream>>>(z, ew, esq, ind_f);

    // 2) gather z_q + per-block squared-diff partial sums
    gather_kernel<<<dim3(ROWS / 256), dim3(256), 0, stream>>>(z, ew, ind_f, zq, block_sums);

    // 3) diff scalar
    finalize_kernel<<<dim3(1), dim3(256), 0, stream>>>(block_sums, diff);
}